// QASAQuantumLayer_53317724013125
// MI455X (gfx1250) — compile-verified
//
#include <hip/hip_runtime.h>
#include <math.h>

typedef float v2f __attribute__((ext_vector_type(2)));
typedef float v8f __attribute__((ext_vector_type(8)));

#define NQ 8
#define DIM 256

// Single-instruction XOR lane exchange within wave32: ds_swizzle_b32 with
// group-of-32 encoding {xor_mask[14:10], or_mask[9:5], and_mask[4:0]}.
template <int MASK>
__device__ inline float swz_xor(float v) {
  const int r = __builtin_amdgcn_ds_swizzle(__float_as_int(v),
                                            (MASK << 10) | 0x1F);
  return __int_as_float(r);
}

// Branchless tanh: tanh(x) = 1 - 2/(e^{2x}+1).  Clamp keeps exp finite; at
// |x|>=15, tanh(x) == +-1 to f32 precision, so the clamp is exact.
__device__ inline float tanh_fast(float x) {
  const float xc = fminf(fmaxf(x, -15.0f), 15.0f);
  const float e  = __expf(2.0f * xc);
  return 1.0f - 2.0f / (e + 1.0f);
}

// ---------------------------------------------------------------------------
// Kernel A: xp = tanh(x @ w_in^T + b_in), plus partial BN sums via f32 atomics.
// One wave per 16-row strip. WMMA f32 16x16x4, 64 K-steps over K=256.
// ---------------------------------------------------------------------------
__global__ void k_gemm_in(const float* __restrict__ x,
                          const float* __restrict__ w_in,
                          const float* __restrict__ b_in,
                          float* __restrict__ xp,
                          float* __restrict__ sums,
                          float* __restrict__ sumsq,
                          int Brows) {
  const int wave = (blockIdx.x * blockDim.x + threadIdx.x) >> 5;
  const int lane = threadIdx.x & 31;
  const int m0 = wave * 16;
  if (m0 >= Brows) return;

  const int nIdx  = lane & 15;          // N (cols) for B, M (rows) for A
  const int khalf = (lane >> 4) * 2;    // 0 or 2: K sub-offset per half-wave
  const bool valid = nIdx < NQ;         // cols 8..15 are zero padding
  const float* arow = x + (size_t)(m0 + nIdx) * DIM + khalf;
  // Clamped row pointer: always in-bounds -> unconditional loads, no exec
  // branching in the inner loop; invalid lanes zeroed with cndmask.
  const float* wrow = w_in + (size_t)(nIdx & 7) * DIM + khalf;

  v8f c = {};
  #pragma unroll 4
  for (int k0 = 0; k0 < DIM; k0 += 4) {
    __builtin_prefetch(arow + k0 + 64, 0, 0);   // global_prefetch for the stream
    v2f a, b;
    a.x = arow[k0 + 0];
    a.y = arow[k0 + 1];
    const float w0 = wrow[k0 + 0];
    const float w1 = wrow[k0 + 1];
    b.x = valid ? w0 : 0.0f;
    b.y = valid ? w1 : 0.0f;
    c = __builtin_amdgcn_wmma_f32_16x16x4_f32(false, a, false, b,
                                              (short)0, c, false, false);
  }

  float s = 0.0f, sq = 0.0f;
  if (valid) {
    const float bn = b_in[nIdx];
    #pragma unroll
    for (int j = 0; j < 8; ++j) {
      const int M = j + (lane >> 4) * 8;
      const float t = tanh_fast(c[j] + bn);
      xp[(size_t)(m0 + M) * NQ + nIdx] = t;
      s  += t;
      sq += t * t;
    }
  }
  s  += swz_xor<16>(s);
  sq += swz_xor<16>(sq);
  if (lane < NQ) {
    atomicAdd(&sums[nIdx], s);
    atomicAdd(&sumsq[nIdx], sq);
  }
}

// ---------------------------------------------------------------------------
// Kernel B: finalize BN scale/shift; precompute cos/sin of weight half-angles.
// ---------------------------------------------------------------------------
__global__ void k_stats(const float* __restrict__ sums,
                        const float* __restrict__ sumsq,
                        const float* __restrict__ gamma,
                        const float* __restrict__ beta,
                        const float* __restrict__ qw,
                        float* __restrict__ scale,
                        float* __restrict__ shift,
                        float* __restrict__ hc,
                        float* __restrict__ hs,
                        float invB) {
  const int t = threadIdx.x;
  if (t < NQ) {
    const float mean = sums[t] * invB;
    const float var  = sumsq[t] * invB - mean * mean;
    const float sc   = gamma[t] * rsqrtf(var + 1e-5f);
    scale[t] = sc;
    shift[t] = beta[t] - mean * sc;
  }
  if (t < 64) {
    const float a = qw[t] * 0.5f;
    hc[t] = cosf(a);
    hs[t] = sinf(a);
  }
}

// ---------------------------------------------------------------------------
// Quantum simulation helpers: 256 complex amps = 8 complex per lane per wave32.
// Amplitude index n = r*32 + lane.  Qubit wire w acts on bit b = 7-w:
// b>=5 -> register pairing (free), b<5 -> cross-lane via ds_swizzle xor.
// ---------------------------------------------------------------------------
__device__ constexpr int safe_shl1(int b) { return (b >= 0) ? (1 << b) : 1; }

template <int B>
__device__ inline void rx_gate(float (&re)[8], float (&im)[8],
                               float c, float s, int lane) {
  // RX(t): new.re = c*self.re + s*other.im ; new.im = c*self.im - s*other.re
  if constexpr (B >= 5) {
    constexpr int m = safe_shl1(B - 5);
    #pragma unroll
    for (int r = 0; r < 8; ++r) {
      if (!(r & m)) {
        const int p = r | m;
        const float r0r = re[r], r0i = im[r], r1r = re[p], r1i = im[p];
        re[r] = c * r0r + s * r1i;  im[r] = c * r0i - s * r1r;
        re[p] = c * r1r + s * r0i;  im[p] = c * r1i - s * r0r;
      }
    }
  } else {
    constexpr int lm = safe_shl1(B);
    (void)lane;
    #pragma unroll
    for (int r = 0; r < 8; ++r) {
      const float pr = swz_xor<lm>(re[r]);
      const float pi = swz_xor<lm>(im[r]);
      re[r] = c * re[r] + s * pi;
      im[r] = c * im[r] - s * pr;
    }
  }
}

template <int B>
__device__ inline void rz_gate(float (&re)[8], float (&im)[8],
                               float c, float s, int lane) {
  // RZ(t): bit==0 -> *(c - i s) ; bit==1 -> *(c + i s)
  #pragma unroll
  for (int r = 0; r < 8; ++r) {
    int bit;
    if constexpr (B >= 5) bit = (r >> (B - 5)) & 1;
    else                  bit = (lane >> B) & 1;
    const float sg = bit ? -s : s;
    const float rr = re[r], ii = im[r];
    re[r] = c * rr + sg * ii;
    im[r] = c * ii - sg * rr;
  }
}

template <int W>
__device__ inline void cnot_gate(float (&re)[8], float (&im)[8], int lane) {
  constexpr int bc = 7 - W;      // control bit
  constexpr int bt = 6 - W;      // target bit
  if constexpr (bt >= 5) {       // W<=1: both bits in register index
    constexpr int mc = safe_shl1(bc - 5);
    constexpr int mt = safe_shl1(bt - 5);
    #pragma unroll
    for (int r = 0; r < 8; ++r) {
      if ((r & mc) && !(r & mt)) {
        const int p = r | mt;
        const float tr = re[r], ti = im[r];
        re[r] = re[p]; im[r] = im[p];
        re[p] = tr;    im[p] = ti;
      }
    }
  } else if constexpr (bc == 5) { // W==2: control reg bit0, target lane bit4
    #pragma unroll
    for (int r = 0; r < 8; ++r) {
      if (r & 1) {
        re[r] = swz_xor<16>(re[r]);
        im[r] = swz_xor<16>(im[r]);
      }
    }
  } else {                        // W>=3: both bits in lane index
    constexpr int lmc = safe_shl1(bc);
    constexpr int lmt = safe_shl1(bt);
    const bool ctl = (lane & lmc) != 0;
    #pragma unroll
    for (int r = 0; r < 8; ++r) {
      const float tr = swz_xor<lmt>(re[r]);
      const float ti = swz_xor<lmt>(im[r]);
      if (ctl) { re[r] = tr; im[r] = ti; }
    }
  }
}

// ---------------------------------------------------------------------------
// Kernel C: one wave per batch element — full 8-qubit circuit + <Z_i>.
// ---------------------------------------------------------------------------
__global__ void k_qsim(const float* __restrict__ xp,
                       const float* __restrict__ scale,
                       const float* __restrict__ shift,
                       const float* __restrict__ hc,
                       const float* __restrict__ hs,
                       float* __restrict__ qv,
                       int Brows) {
  const int wave = (blockIdx.x * blockDim.x + threadIdx.x) >> 5;
  const int lane = threadIdx.x & 31;
  if (wave >= Brows) return;

  float re[8], im[8];
  #pragma unroll
  for (int r = 0; r < 8; ++r) { re[r] = 0.0f; im[r] = 0.0f; }
  if (lane == 0) re[0] = 1.0f;   // |0...0>

  // Input encoding: RX(ang_i) then RZ(ang_i) on wire i (bit 7-i).
  #define ENC(i)                                                             \
    do {                                                                     \
      const float ang = 0.5f * (xp[(size_t)wave * NQ + (i)] * scale[(i)]     \
                                + shift[(i)]);                               \
      const float cc = cosf(ang), ss = sinf(ang);                            \
      rx_gate<7 - (i)>(re, im, cc, ss, lane);                                \
      rz_gate<7 - (i)>(re, im, cc, ss, lane);                                \
    } while (0)
  ENC(0); ENC(1); ENC(2); ENC(3); ENC(4); ENC(5); ENC(6); ENC(7);
  #undef ENC

  // Trainable layers: 4 x [ per-qubit RX,RZ ; CNOT ladder ].
  for (int l = 0; l < 4; ++l) {
    const float* lc = hc + l * 16;
    const float* ls = hs + l * 16;
    #define LQ(i)                                                            \
      do {                                                                   \
        rx_gate<7 - (i)>(re, im, lc[2 * (i)],     ls[2 * (i)],     lane);    \
        rz_gate<7 - (i)>(re, im, lc[2 * (i) + 1], ls[2 * (i) + 1], lane);    \
      } while (0)
    LQ(0); LQ(1); LQ(2); LQ(3); LQ(4); LQ(5); LQ(6); LQ(7);
    #undef LQ
    cnot_gate<0>(re, im, lane);
    cnot_gate<1>(re, im, lane);
    cnot_gate<2>(re, im, lane);
    cnot_gate<3>(re, im, lane);
    cnot_gate<4>(re, im, lane);
    cnot_gate<5>(re, im, lane);
    cnot_gate<6>(re, im, lane);
  }

  // <Z_i> = sum_n |amp_n|^2 * (1 - 2*bit_{7-i}(n))
  float ez[8];
  #pragma unroll
  for (int i = 0; i < 8; ++i) ez[i] = 0.0f;
  #pragma unroll
  for (int r = 0; r < 8; ++r) {
    const float p = re[r] * re[r] + im[r] * im[r];
    #pragma unroll
    for (int i = 0; i < 8; ++i) {
      const int b = 7 - i;
      const int bit = (b >= 5) ? ((r >> (b - 5)) & 1) : ((lane >> b) & 1);
      ez[i] += bit ? -p : p;
    }
  }
  #pragma unroll
  for (int i = 0; i < 8; ++i) {
    ez[i] += swz_xor<16>(ez[i]);
    ez[i] += swz_xor<8>(ez[i]);
    ez[i] += swz_xor<4>(ez[i]);
    ez[i] += swz_xor<2>(ez[i]);
    ez[i] += swz_xor<1>(ez[i]);
  }
  if (lane == 0) {
    #pragma unroll
    for (int i = 0; i < 8; ++i) qv[(size_t)wave * NQ + i] = ez[i];
  }
}

// ---------------------------------------------------------------------------
// Kernel D: out = x + q @ w_out^T + b_out.  One wave per 16x16 output tile,
// two WMMA f32 16x16x4 steps (K=8).
// ---------------------------------------------------------------------------
__global__ void k_out(const float* __restrict__ x,
                      const float* __restrict__ qv,
                      const float* __restrict__ w_out,
                      const float* __restrict__ b_out,
                      float* __restrict__ out,
                      int Brows) {
  const int wave = (blockIdx.x * blockDim.x + threadIdx.x) >> 5;
  const int lane = threadIdx.x & 31;
  const int m0 = (wave >> 4) * 16;        // 16 column tiles across DIM=256
  const int n0 = (wave & 15) * 16;
  if (m0 >= Brows) return;

  const int sel   = lane & 15;
  const int khalf = (lane >> 4) * 2;

  v2f a0, a1, b0, b1;
  const float* qr = qv + (size_t)(m0 + sel) * NQ;
  a0.x = qr[khalf + 0];  a0.y = qr[khalf + 1];
  a1.x = qr[khalf + 4];  a1.y = qr[khalf + 5];
  const float* wr = w_out + (size_t)(n0 + sel) * NQ;   // B = w_out^T
  b0.x = wr[khalf + 0];  b0.y = wr[khalf + 1];
  b1.x = wr[khalf + 4];  b1.y = wr[khalf + 5];

  v8f c = {};
  c = __builtin_amdgcn_wmma_f32_16x16x4_f32(false, a0, false, b0,
                                            (short)0, c, false, false);
  c = __builtin_amdgcn_wmma_f32_16x16x4_f32(false, a1, false, b1,
                                            (short)0, c, false, false);

  const float bo = b_out[n0 + sel];
  #pragma unroll
  for (int j = 0; j < 8; ++j) {
    const int M = j + (lane >> 4) * 8;
    const size_t idx = (size_t)(m0 + M) * DIM + n0 + sel;
    out[idx] = x[idx] + c[j] + bo;
  }
}

// ---------------------------------------------------------------------------
extern "C" void kernel_launch(void* const* d_in, const int* in_sizes, int n_in,
                              void* d_out, int out_size, void* d_ws, size_t ws_size,
                              hipStream_t stream) {
  const float* x     = (const float*)d_in[0];   // [B,256]
  const float* w_in  = (const float*)d_in[1];   // [8,256]
  const float* b_in  = (const float*)d_in[2];   // [8]
  const float* gamma = (const float*)d_in[3];   // [8]
  const float* beta  = (const float*)d_in[4];   // [8]
  const float* qw    = (const float*)d_in[5];   // [4,16]
  const float* w_out = (const float*)d_in[6];   // [256,8]
  const float* b_out = (const float*)d_in[7];   // [256]
  float* out = (float*)d_out;

  const int Brows = in_sizes[0] / DIM;          // 32768

  // Workspace layout (floats)
  float* sums  = (float*)d_ws;       // [8]
  float* sumsq = sums + 8;           // [8]
  float* scale = sums + 16;          // [8]
  float* shift = sums + 24;          // [8]
  float* hc    = sums + 32;          // [64]
  float* hs    = sums + 96;          // [64]
  float* xp    = sums + 160;         // [B,8]
  float* qv    = xp + (size_t)Brows * NQ;  // [B,8]

  hipMemsetAsync(sums, 0, 16 * sizeof(float), stream);   // zero BN accumulators

  // Kernel A: waves = B/16, 8 waves per block
  {
    const int waves = Brows / 16;
    const int blocks = (waves * 32 + 255) / 256;
    k_gemm_in<<<blocks, 256, 0, stream>>>(x, w_in, b_in, xp, sums, sumsq, Brows);
  }
  // Kernel B: finalize stats + weight trig tables
  k_stats<<<1, 64, 0, stream>>>(sums, sumsq, gamma, beta, qw,
                                scale, shift, hc, hs, 1.0f / (float)Brows);
  // Kernel C: one wave per element
  {
    const int blocks = (Brows * 32 + 255) / 256;
    k_qsim<<<blocks, 256, 0, stream>>>(xp, scale, shift, hc, hs, qv, Brows);
  }
  // Kernel D: one wave per 16x16 tile, (B/16)*16 tiles
  {
    const int waves = (Brows / 16) * (DIM / 16);
    const int blocks = (waves * 32 + 255) / 256;
    k_out<<<blocks, 256, 0, stream>>>(x, qv, w_out, b_out, out, Brows);
  }
}